// EmbedGNN_64888365908124
// MI455X (gfx1250) — compile-verified
//
#include <hip/hip_runtime.h>

// ---------------------------------------------------------------------------
// EmbedGNN for MI455X (gfx1250, wave32).
// All GEMM work uses V_WMMA_F32_16X16X4_F32 (D = A(16x4) * B(4x16) + C).
//
// Fragment layouts (CDNA5 ISA 7.12.2, wave32):
//   A (16x4 f32, 2 VGPR = v2f/lane): lanes 0-15 hold M=lane, {K=0,K=1};
//                                    lanes 16-31 hold M=lane-16, {K=2,K=3}.
//   C/D (16x16 f32, 8 VGPR = v8f):   element r of lane L is
//                                    (M = r + 8*(L>>4), N = L&15).
//   B (4x16 f32, 2 VGPR): mirrors A's K striping:
//                                    lanes 0-15 {K=0,K=1}, lanes 16-31 {K=2,K=3},
//                                    N = L&15 within the VGPR.
//
// Node arrays (25.6 MB each) are L2-resident (192 MB L2); the optimization
// target is L2 bandwidth + atomic throughput, so node-GEMM A tiles are staged
// in LDS (rows padded to 132 floats -> conflict-free ds_load_b64 fragments).
// ---------------------------------------------------------------------------

typedef float v2f __attribute__((ext_vector_type(2)));
typedef float v8f __attribute__((ext_vector_type(8)));

#define LATENT 128
#define EFEAT  16
#define LDS_PITCH 132   // 132 mod 64 == 4 -> fragment reads hit all 64 banks

__device__ __forceinline__ v8f wmma_f32_k4(v2f a, v2f b, v8f c) {
  return __builtin_amdgcn_wmma_f32_16x16x4_f32(
      /*neg_a=*/false, a, /*neg_b=*/false, b,
      /*c_mod=*/(short)0, c, /*reuse_a=*/false, /*reuse_b=*/false);
}

__device__ __forceinline__ void atomic_add_f32(float* p, float v) {
  // Non-returning HW f32 atomic -> global_atomic_add_f32 (STOREcnt path).
  unsafeAtomicAdd(p, v);
}

// Cooperative copy of a 16x128 f32 tile (contiguous rows) into padded LDS.
__device__ __forceinline__ void stage_tile(float (*dst)[LDS_PITCH],
                                           const float* __restrict__ src,
                                           int tid) {
  const float2* s2 = (const float2*)src;   // 1024 float2 per tile
#pragma unroll
  for (int i = 0; i < 4; ++i) {
    const int idx = i * 256 + tid;         // 0..1023
    const int row = idx >> 6;              // 64 float2 per row
    const int c2  = idx & 63;
    *(float2*)&dst[row][c2 * 2] = s2[idx];
  }
}

// ---------------------------------------------------------------------------
// K1: fused edge MLP + relu + scatter-add onto nodes.
//     Per wave: a 16-edge tile. A = edge_feat[16x16], B = W_e2l[16x128]
//     (8 N-tiles x 4 K-steps of WMMA), then 128 atomics per edge.
// ---------------------------------------------------------------------------
__global__ __launch_bounds__(256) void k_edge_embed_scatter(
    const float* __restrict__ edge_feat, const float* __restrict__ W_e2l,
    const int* __restrict__ e2n_dst, float* __restrict__ pool,
    int n_tiles, int total_waves) {
  const int lane    = threadIdx.x & 31;
  const int lane_lo = lane & 15;
  const int lane_hi = lane >> 4;
  const int wave    = (int)((blockIdx.x * blockDim.x + threadIdx.x) >> 5);

  // Hoist all B fragments of W_e2l (16x128) into registers: shared by all tiles.
  v2f Bf[8][4];
#pragma unroll
  for (int nt = 0; nt < 8; ++nt) {
#pragma unroll
    for (int ks = 0; ks < 4; ++ks) {
      const int k0  = 4 * ks + 2 * lane_hi;
      const int col = nt * 16 + lane_lo;
      Bf[nt][ks].x = W_e2l[(k0 + 0) * LATENT + col];
      Bf[nt][ks].y = W_e2l[(k0 + 1) * LATENT + col];
    }
  }

  for (int t = wave; t < n_tiles; t += total_waves) {
    const int e0 = t * 16;
    // Prefetch the next tile this wave will touch; keep it near (WGP scope).
    __builtin_prefetch(edge_feat + (size_t)(e0 + 16 * total_waves) * EFEAT, 0, 3);

    const float* arow = edge_feat + (size_t)(e0 + lane_lo) * EFEAT + 2 * lane_hi;
    v2f a[4];
#pragma unroll
    for (int ks = 0; ks < 4; ++ks) a[ks] = *(const v2f*)(arow + 4 * ks);

    int dstm[8];
#pragma unroll
    for (int r = 0; r < 8; ++r) dstm[r] = e2n_dst[e0 + r + 8 * lane_hi];

#pragma unroll
    for (int nt = 0; nt < 8; ++nt) {
      v8f acc = {0.f, 0.f, 0.f, 0.f, 0.f, 0.f, 0.f, 0.f};
#pragma unroll
      for (int ks = 0; ks < 4; ++ks) acc = wmma_f32_k4(a[ks], Bf[nt][ks], acc);
      float* pcol = pool + (nt * 16 + lane_lo);
#pragma unroll
      for (int r = 0; r < 8; ++r) {
        const float v = fmaxf(acc[r], 0.f);  // relu before pooling
        atomic_add_f32(pcol + (size_t)dstm[r] * LATENT, v);
      }
    }
  }
}

// ---------------------------------------------------------------------------
// K2: static_message = e2n_pool @ W0 ; h = relu(static_message)
//     Block: 16-row tile staged in LDS; wave nt covers 16 columns.
// ---------------------------------------------------------------------------
__global__ __launch_bounds__(256) void k_static_gemm(
    const float* __restrict__ X, const float* __restrict__ W0,
    float* __restrict__ static_msg, float* __restrict__ h) {
  __shared__ float sX[16][LDS_PITCH];
  const int tid     = threadIdx.x;
  const int lane    = tid & 31;
  const int lane_lo = lane & 15;
  const int lane_hi = lane >> 4;
  const int nt      = tid >> 5;             // 0..7 -> column tile
  const int m0      = blockIdx.x * 16;
  const int col     = nt * 16 + lane_lo;

  stage_tile(sX, X + (size_t)m0 * LATENT, tid);
  __syncthreads();

  v8f acc = {0.f, 0.f, 0.f, 0.f, 0.f, 0.f, 0.f, 0.f};
#pragma unroll 8
  for (int ks = 0; ks < 32; ++ks) {
    const int k0 = 4 * ks + 2 * lane_hi;
    const v2f a  = *(const v2f*)&sX[lane_lo][k0];
    v2f b;
    b.x = W0[(k0 + 0) * LATENT + col];
    b.y = W0[(k0 + 1) * LATENT + col];
    acc = wmma_f32_k4(a, b, acc);
  }
#pragma unroll
  for (int r = 0; r < 8; ++r) {
    const size_t o = (size_t)(m0 + r + 8 * lane_hi) * LATENT + col;
    const float  v = acc[r];
    static_msg[o] = v;
    h[o]          = fmaxf(v, 0.f);
  }
}

// ---------------------------------------------------------------------------
// K3: n2n spmm: pool[dst] += h[src] over the edge list.
//     One lane per (edge, 4-float chunk); a wave covers one 512B node row.
// ---------------------------------------------------------------------------
__global__ __launch_bounds__(256) void k_n2n_scatter(
    const float* __restrict__ H, const int* __restrict__ esrc,
    const int* __restrict__ edst, float* __restrict__ pool, int n_edges) {
  const long long idx = (long long)blockIdx.x * blockDim.x + threadIdx.x;
  const int e = (int)(idx >> 5);
  if (e >= n_edges) return;
  const int c = (int)(idx & 31);
  const int s = esrc[e];
  const int d = edst[e];
  const float4 v = *(const float4*)(H + (size_t)s * LATENT + c * 4);
  float* p = pool + (size_t)d * LATENT + c * 4;
  atomic_add_f32(p + 0, v.x);
  atomic_add_f32(p + 1, v.y);
  atomic_add_f32(p + 2, v.z);
  atomic_add_f32(p + 3, v.w);
}

// ---------------------------------------------------------------------------
// K4: h_out = relu(static_msg + H @ W1 + P @ W2)
//     Both A tiles staged in LDS; two chained WMMA accumulation passes.
// ---------------------------------------------------------------------------
__global__ __launch_bounds__(256) void k_merge_gemm(
    const float* __restrict__ H, const float* __restrict__ P,
    const float* __restrict__ W1, const float* __restrict__ W2,
    const float* __restrict__ static_msg, float* __restrict__ Hout) {
  __shared__ float sH[16][LDS_PITCH];
  __shared__ float sP[16][LDS_PITCH];
  const int tid     = threadIdx.x;
  const int lane    = tid & 31;
  const int lane_lo = lane & 15;
  const int lane_hi = lane >> 4;
  const int nt      = tid >> 5;
  const int m0      = blockIdx.x * 16;
  const int col     = nt * 16 + lane_lo;

  stage_tile(sH, H + (size_t)m0 * LATENT, tid);
  stage_tile(sP, P + (size_t)m0 * LATENT, tid);
  __syncthreads();

  v8f acc = {0.f, 0.f, 0.f, 0.f, 0.f, 0.f, 0.f, 0.f};
#pragma unroll 8
  for (int ks = 0; ks < 32; ++ks) {
    const int k0 = 4 * ks + 2 * lane_hi;
    const v2f a  = *(const v2f*)&sH[lane_lo][k0];
    v2f b;
    b.x = W1[(k0 + 0) * LATENT + col];
    b.y = W1[(k0 + 1) * LATENT + col];
    acc = wmma_f32_k4(a, b, acc);
  }
#pragma unroll 8
  for (int ks = 0; ks < 32; ++ks) {
    const int k0 = 4 * ks + 2 * lane_hi;
    const v2f a  = *(const v2f*)&sP[lane_lo][k0];
    v2f b;
    b.x = W2[(k0 + 0) * LATENT + col];
    b.y = W2[(k0 + 1) * LATENT + col];
    acc = wmma_f32_k4(a, b, acc);
  }

#pragma unroll
  for (int r = 0; r < 8; ++r) {
    const size_t o = (size_t)(m0 + r + 8 * lane_hi) * LATENT + col;
    Hout[o] = fmaxf(static_msg[o] + acc[r], 0.f);
  }
}

// ---------------------------------------------------------------------------
extern "C" void kernel_launch(void* const* d_in, const int* in_sizes, int n_in,
                              void* d_out, int out_size, void* d_ws, size_t ws_size,
                              hipStream_t stream) {
  const float* edge_feat = (const float*)d_in[0];
  const float* W_e2l     = (const float*)d_in[1];
  const float* W0        = (const float*)d_in[2];
  const float* W1s[3]    = {(const float*)d_in[3], (const float*)d_in[5], (const float*)d_in[7]};
  const float* W2s[3]    = {(const float*)d_in[4], (const float*)d_in[6], (const float*)d_in[8]};
  const int*   e2n_dst   = (const int*)d_in[9];
  const int*   edge_src  = (const int*)d_in[10];
  const int*   edge_dst  = (const int*)d_in[11];

  const int E = in_sizes[9];          // 1,600,000 (multiple of 16)
  const int N = out_size / LATENT;    // 50,000   (multiple of 16)
  const size_t node_elems = (size_t)N * LATENT;

  float* ws         = (float*)d_ws;
  float* e2n_pool   = ws + 0 * node_elems;
  float* static_msg = ws + 1 * node_elems;
  float* n2n_pool   = ws + 2 * node_elems;
  float* hA         = ws + 3 * node_elems;
  float* hB         = ws + 4 * node_elems;
  float* hOut       = (float*)d_out;

  // ---- edge MLP + pool ----------------------------------------------------
  hipMemsetAsync(e2n_pool, 0, node_elems * sizeof(float), stream);
  const int k1_blocks = 4096;                       // 32768 waves, grid-stride
  k_edge_embed_scatter<<<k1_blocks, 256, 0, stream>>>(
      edge_feat, W_e2l, e2n_dst, e2n_pool, E / 16, k1_blocks * 8);

  // ---- static message + initial h ----------------------------------------
  k_static_gemm<<<N / 16, 256, 0, stream>>>(e2n_pool, W0, static_msg, hA);

  // ---- 3 message-passing rounds ------------------------------------------
  const float* hin = hA;
  float* houts[3]  = {hB, hA, hOut};
  const int k3_blocks = (int)(((long long)E * 32 + 255) / 256);
  for (int it = 0; it < 3; ++it) {
    hipMemsetAsync(n2n_pool, 0, node_elems * sizeof(float), stream);
    k_n2n_scatter<<<k3_blocks, 256, 0, stream>>>(hin, edge_src, edge_dst, n2n_pool, E);
    k_merge_gemm<<<N / 16, 256, 0, stream>>>(hin, n2n_pool, W1s[it], W2s[it],
                                             static_msg, houts[it]);
    hin = houts[it];
  }
}